// iVGAE_Encoder_57604101373963
// MI455X (gfx1250) — compile-verified
//
#include <hip/hip_runtime.h>
#include <hip/hip_bf16.h>

typedef float v2f __attribute__((ext_vector_type(2)));
typedef float v8f __attribute__((ext_vector_type(8)));

#define N_NODES 100000
#define IN_C 128
#define HID_C 64
#define OUT_C 32

// ---------------------------------------------------------------------------
// generic zero-fill
// ---------------------------------------------------------------------------
__global__ void zero_kernel(float* __restrict__ p, size_t n) {
    size_t i = (size_t)blockIdx.x * blockDim.x + threadIdx.x;
    if (i < n) p[i] = 0.0f;
}

// ---------------------------------------------------------------------------
// degree scatter: deg[dst[e]] += 1
// ---------------------------------------------------------------------------
__global__ void degree_kernel(const long long* __restrict__ dst,
                              float* __restrict__ deg, int E) {
    int e = blockIdx.x * blockDim.x + threadIdx.x;
    if (e < E) atomicAdd(&deg[(int)dst[e]], 1.0f);
}

// dinv[i] = rsqrt(deg[i] + 1)   (in-place over deg)
__global__ void dinv_kernel(float* __restrict__ deg, int n) {
    int i = blockIdx.x * blockDim.x + threadIdx.x;
    if (i < n) deg[i] = rsqrtf(deg[i] + 1.0f);
}

// ---------------------------------------------------------------------------
// f32 WMMA GEMM: C[M,N] = A[M,K] @ B[K,N], one wave per 16x16 tile,
// V_WMMA_F32_16X16X4_F32 accumulating K in steps of 4.
// A frag: lane<16 -> K=k0,k0+1 of row (lane&15); lane>=16 -> K=k0+2,k0+3
// B frag: lane<16 -> rows k0,k0+1 at col (lane&15); lane>=16 -> rows k0+2,k0+3
// C/D   : VGPR r, row = 16*tm + r + (lane>=16)*8, col = 16*tn + (lane&15)
// ---------------------------------------------------------------------------
__global__ void wmma_gemm_f32(const float* __restrict__ A,
                              const float* __restrict__ B,
                              float* __restrict__ C,
                              int M, int K, int N) {
    const int wavesPerBlock = blockDim.x >> 5;
    const int waveId = blockIdx.x * wavesPerBlock + (threadIdx.x >> 5);
    const int tilesN = N >> 4;
    const int numTiles = (M >> 4) * tilesN;
    if (waveId >= numTiles) return;          // whole wave exits: EXEC stays all-1
    const int tm = waveId / tilesN;
    const int tn = waveId % tilesN;
    const int lane = threadIdx.x & 31;
    const int half = lane >> 4;              // 0 or 1
    const int l16  = lane & 15;

    const int arow = tm * 16 + l16;
    const int bcol = tn * 16 + l16;

    v8f acc = {};
    for (int k0 = 0; k0 < K; k0 += 4) {
        const int ka = k0 + half * 2;
        v2f a = *(const v2f*)(A + (size_t)arow * K + ka);     // A[arow, ka..ka+1]
        v2f b;
        b.x = B[(size_t)(ka)     * N + bcol];                 // B[ka,   bcol]
        b.y = B[(size_t)(ka + 1) * N + bcol];                 // B[ka+1, bcol]
        acc = __builtin_amdgcn_wmma_f32_16x16x4_f32(
            /*neg_a=*/false, a, /*neg_b=*/false, b,
            /*c_mod=*/(short)0, acc, /*reuse_a=*/false, /*reuse_b=*/false);
    }
#pragma unroll
    for (int r = 0; r < 8; ++r) {
        const int row = tm * 16 + r + half * 8;
        C[(size_t)row * N + tn * 16 + l16] = acc[r];
    }
}

// ---------------------------------------------------------------------------
// layer-1 edge scatter: one wave per edge, 64 channels (2 per lane, float2)
// agg[dst] += h[src] * dinv[src]*dinv[dst]
// ---------------------------------------------------------------------------
__global__ void scatter64_kernel(const long long* __restrict__ src,
                                 const long long* __restrict__ dst,
                                 const float* __restrict__ dinv,
                                 const float* __restrict__ h,
                                 float* __restrict__ agg, int E) {
    const int wid = (blockIdx.x * blockDim.x + threadIdx.x) >> 5;
    const int lane = threadIdx.x & 31;
    if (wid >= E) return;
    const int s = (int)src[wid];
    const int d = (int)dst[wid];
    const float c = dinv[s] * dinv[d];
    const v2f hv = *(const v2f*)(h + (size_t)s * HID_C + lane * 2);
    atomicAdd(&agg[(size_t)d * HID_C + lane * 2],     hv.x * c);
    atomicAdd(&agg[(size_t)d * HID_C + lane * 2 + 1], hv.y * c);
}

// h = relu(agg + h_lin * dinv_i^2 + b0)   (written in place over agg)
__global__ void finalize1_kernel(float* __restrict__ agg,
                                 const float* __restrict__ hlin,
                                 const float* __restrict__ dinv,
                                 const float* __restrict__ b0, int n) {
    int i = blockIdx.x * blockDim.x + threadIdx.x;
    if (i < n) {
        const int node = i >> 6;       // /64
        const int ch   = i & 63;
        const float dv = dinv[node];
        float v = agg[i] + hlin[i] * dv * dv + b0[ch];
        agg[i] = v > 0.0f ? v : 0.0f;
    }
}

// ---------------------------------------------------------------------------
// fused layer-2/3 edge scatter: one wave per edge, 32 channels, 1 per lane,
// two outputs (mean, logstd) sharing edge reads + coef
// ---------------------------------------------------------------------------
__global__ void scatter32_fused_kernel(const long long* __restrict__ src,
                                       const long long* __restrict__ dst,
                                       const float* __restrict__ dinv,
                                       const float* __restrict__ hm,
                                       const float* __restrict__ hl,
                                       float* __restrict__ om,
                                       float* __restrict__ ol, int E) {
    const int wid = (blockIdx.x * blockDim.x + threadIdx.x) >> 5;
    const int lane = threadIdx.x & 31;
    if (wid >= E) return;
    const int s = (int)src[wid];
    const int d = (int)dst[wid];
    const float c = dinv[s] * dinv[d];
    const float vm = hm[(size_t)s * OUT_C + lane] * c;
    const float vl = hl[(size_t)s * OUT_C + lane] * c;
    atomicAdd(&om[(size_t)d * OUT_C + lane], vm);
    atomicAdd(&ol[(size_t)d * OUT_C + lane], vl);
}

// out += h_lin * dinv_i^2 + bias  (both outputs)
__global__ void finalize2_kernel(float* __restrict__ om, float* __restrict__ ol,
                                 const float* __restrict__ hm,
                                 const float* __restrict__ hl,
                                 const float* __restrict__ dinv,
                                 const float* __restrict__ bm,
                                 const float* __restrict__ bl, int n) {
    int i = blockIdx.x * blockDim.x + threadIdx.x;
    if (i < n) {
        const int node = i >> 5;       // /32
        const int ch   = i & 31;
        const float dv = dinv[node];
        const float d2 = dv * dv;
        om[i] += hm[i] * d2 + bm[ch];
        ol[i] += hl[i] * d2 + bl[ch];
    }
}

// ---------------------------------------------------------------------------
// launcher
// ---------------------------------------------------------------------------
extern "C" void kernel_launch(void* const* d_in, const int* in_sizes, int n_in,
                              void* d_out, int out_size, void* d_ws, size_t ws_size,
                              hipStream_t stream) {
    const float*     x   = (const float*)d_in[0];
    const long long* ei  = (const long long*)d_in[1];
    const float*     W0  = (const float*)d_in[2];
    const float*     b0  = (const float*)d_in[3];
    const float*     Wm  = (const float*)d_in[4];
    const float*     bm  = (const float*)d_in[5];
    const float*     Wl  = (const float*)d_in[6];
    const float*     bl  = (const float*)d_in[7];

    const int M = in_sizes[0] / IN_C;     // 100000 nodes
    const int E = in_sizes[1] / 2;        // 1600000 edges
    const long long* src = ei;
    const long long* dst = ei + E;

    // workspace layout
    float* dinv = (float*)d_ws;                       // M floats (deg -> dinv)
    float* bufA = dinv + M;                           // M*64 (h_lin, then hm||hl)
    float* bufB = bufA + (size_t)M * HID_C;           // M*64 (agg1 -> h)
    float* hm   = bufA;                               // M*32
    float* hl   = bufA + (size_t)M * OUT_C;           // M*32
    float* om   = (float*)d_out;                      // M*32 (mean)
    float* ol   = om + (size_t)M * OUT_C;             // M*32 (logstd)

    const int TPB = 256;

    // 1) degrees -> dinv
    zero_kernel<<<(M + TPB - 1) / TPB, TPB, 0, stream>>>(dinv, (size_t)M);
    degree_kernel<<<(E + TPB - 1) / TPB, TPB, 0, stream>>>(dst, dinv, E);
    dinv_kernel<<<(M + TPB - 1) / TPB, TPB, 0, stream>>>(dinv, M);

    // 2) h_lin = x @ W0   (WMMA f32, one wave per 16x16 tile)
    {
        int tiles = (M / 16) * (HID_C / 16);
        wmma_gemm_f32<<<(tiles + 7) / 8, TPB, 0, stream>>>(x, W0, bufA, M, IN_C, HID_C);
    }

    // 3) agg1 = scatter(h_lin[src]*coef -> dst)
    zero_kernel<<<(int)(((size_t)M * HID_C + TPB - 1) / TPB), TPB, 0, stream>>>(bufB, (size_t)M * HID_C);
    scatter64_kernel<<<(E * 32 + TPB - 1) / TPB, TPB, 0, stream>>>(src, dst, dinv, bufA, bufB, E);

    // 4) h = relu(agg1 + h_lin*dinv^2 + b0)
    finalize1_kernel<<<(int)(((size_t)M * HID_C + TPB - 1) / TPB), TPB, 0, stream>>>(
        bufB, bufA, dinv, b0, M * HID_C);

    // 5) hm = h @ Wm ; hl = h @ Wl   (bufA reused)
    {
        int tiles = (M / 16) * (OUT_C / 16);
        wmma_gemm_f32<<<(tiles + 7) / 8, TPB, 0, stream>>>(bufB, Wm, hm, M, HID_C, OUT_C);
        wmma_gemm_f32<<<(tiles + 7) / 8, TPB, 0, stream>>>(bufB, Wl, hl, M, HID_C, OUT_C);
    }

    // 6) zero outputs, fused edge scatter for mean & logstd
    zero_kernel<<<(int)(((size_t)M * OUT_C * 2 + TPB - 1) / TPB), TPB, 0, stream>>>(
        om, (size_t)M * OUT_C * 2);
    scatter32_fused_kernel<<<(E * 32 + TPB - 1) / TPB, TPB, 0, stream>>>(
        src, dst, dinv, hm, hl, om, ol, E);

    // 7) self-loop + bias
    finalize2_kernel<<<(int)(((size_t)M * OUT_C + TPB - 1) / TPB), TPB, 0, stream>>>(
        om, ol, hm, hl, dinv, bm, bl, M * OUT_C);
}